// CustomMaxAbsPool2d_54863912239744
// MI455X (gfx1250) — compile-verified
//
#include <hip/hip_runtime.h>
#include <stdint.h>

// ---------------------------------------------------------------------------
// CustomMaxAbsPool2d for x[16,64,256,256] f32, K=S=2.
//
// Memory-bound streaming kernel (335 MB @ 23.3 TB/s ~= 14.4 us floor).
// CDNA5 async-to-LDS DMA path (global_load_async_to_lds_b128, th:TH_LOAD_NT)
// with double buffering, ds_load_b128 compute reads, and non-temporal b128
// output stores. VALU/byte ratio leaves >4x VALU headroom -> pure BW bound.
//
// Tiling: one tile = 16 input rows x 256 cols (16 KB) -> 8 out rows x 128
// (4 KB). Flat tile index g: input offset = g*4096 floats, output offset =
// g*1024 floats (both exactly linear -> trivially coalesced).
// ---------------------------------------------------------------------------

typedef __attribute__((ext_vector_type(4))) float v4f;

#define THREADS          256
#define TILES_PER_BLOCK  8
#define TILE_IN_FLOATS   4096   // 16 rows * 256 cols
#define TILE_OUT_FLOATS  1024   // 8 rows * 128 cols

// One 128-bit async DMA transfer: global -> LDS. GV addressing (64-bit VGPR
// address pair, SADDR=off). VDST carries the wave-relative LDS byte address
// (low 32 bits of the generic pointer to a __shared__ object). Non-temporal:
// the 268 MB input is read exactly once, don't pollute WGP$/L2.
__device__ __forceinline__ void async_load_b128(uint32_t lds_byte, const void* gaddr) {
  asm volatile("global_load_async_to_lds_b128 %0, %1, off th:TH_LOAD_NT"
               :: "v"(lds_byte), "v"(gaddr)
               : "memory");
}

// One 2x2 window: m = max|.|, mask ties, max of masked values.
// (Reference multiplies by a 0/1 mask; select against +0.0f is numerically
// identical, differing only in sign-of-zero.)
__device__ __forceinline__ float win2x2(float a, float b, float c, float d) {
  float aa = __builtin_fabsf(a), ab = __builtin_fabsf(b);
  float ac = __builtin_fabsf(c), ad = __builtin_fabsf(d);
  float m  = fmaxf(fmaxf(aa, ab), fmaxf(ac, ad));
  float pa = (aa == m) ? a : 0.0f;
  float pb = (ab == m) ? b : 0.0f;
  float pc = (ac == m) ? c : 0.0f;
  float pd = (ad == m) ? d : 0.0f;
  return fmaxf(fmaxf(pa, pb), fmaxf(pc, pd));
}

__global__ __launch_bounds__(THREADS)
void maxabspool2d_kernel(const float* __restrict__ x, float* __restrict__ out) {
  __shared__ __align__(16) float lds[2][TILE_IN_FLOATS];   // 2 x 16 KB double buffer

  const int tid = threadIdx.x;
  // Wave-relative LDS byte addresses of the two buffers (low 32 bits of the
  // generic pointer are the LDS offset in CDNA5's aperture scheme).
  const uint32_t ldsb[2] = {
    (uint32_t)(uintptr_t)(void*)&lds[0][0],
    (uint32_t)(uintptr_t)(void*)&lds[1][0],
  };

  const long g0 = (long)blockIdx.x * TILES_PER_BLOCK;

  // Issue the 16 KB DMA for tile (g0+i) into buffer (i&1): 256 threads x
  // 4 x 16 B, perfectly coalesced 128-bit transfers. 4 async instructions
  // per wave per tile -> ASYNCcnt bookkeeping below.
  auto prefetch = [&](int i) {
    const char* src = (const char*)(x + ((g0 + i) << 12));  // *4096 floats
    const uint32_t lb = ldsb[i & 1];
#pragma unroll
    for (int k = 0; k < 4; ++k) {
      const uint32_t off = (uint32_t)(tid * 16 + k * 4096);
      async_load_b128(lb + off, src + off);
    }
  };

  prefetch(0);

  const int r = tid >> 5;   // output row within tile   [0,8)
  const int q = tid & 31;   // output col quad          [0,32)

#pragma unroll
  for (int i = 0; i < TILES_PER_BLOCK; ++i) {
    if (i + 1 < TILES_PER_BLOCK) {
      prefetch(i + 1);                                   // overlap DMA w/ compute
      // Async loads complete in order: <=4 outstanding => tile i is in LDS.
      asm volatile("s_wait_asynccnt 0x4" ::: "memory");
    } else {
      asm volatile("s_wait_asynccnt 0x0" ::: "memory");
    }
    __syncthreads();   // all waves' DMA for tile i visible

    const float* L  = lds[i & 1];
    const float* p0 = L + (r * 512 + q * 8);             // input row 2r, col 8q
    v4f x0 = *(const v4f*)(p0);
    v4f x1 = *(const v4f*)(p0 + 4);
    v4f y0 = *(const v4f*)(p0 + 256);                    // input row 2r+1
    v4f y1 = *(const v4f*)(p0 + 260);

    v4f o;
    o.x = win2x2(x0.x, x0.y, y0.x, y0.y);
    o.y = win2x2(x0.z, x0.w, y0.z, y0.w);
    o.z = win2x2(x1.x, x1.y, y1.x, y1.y);
    o.w = win2x2(x1.z, x1.w, y1.z, y1.w);

    // out offset: tile*1024 + r*128 + 4q == tile*1024 + tid*4 ; write-once
    // stream -> non-temporal 128-bit store.
    float* d = out + ((g0 + i) << 10) + (tid << 2);
    __builtin_nontemporal_store(o, (v4f*)d);

    __syncthreads();   // buffer (i&1) free for reuse at prefetch(i+2)
  }
}

extern "C" void kernel_launch(void* const* d_in, const int* in_sizes, int n_in,
                              void* d_out, int out_size, void* d_ws, size_t ws_size,
                              hipStream_t stream) {
  const float* x = (const float*)d_in[0];
  float* out = (float*)d_out;

  const long n_elems = (long)in_sizes[0];              // 16*64*256*256 = 67108864
  const long tiles   = n_elems / TILE_IN_FLOATS;       // 16384
  const int  blocks  = (int)(tiles / TILES_PER_BLOCK); // 2048

  maxabspool2d_kernel<<<blocks, THREADS, 0, stream>>>(x, out);
}